// ChamferDistance_21406117004171
// MI455X (gfx1250) — compile-verified
//
#include <hip/hip_runtime.h>

typedef __attribute__((ext_vector_type(2))) float v2f;
typedef __attribute__((ext_vector_type(8))) float v8f;

// min without fminf's sNaN-canonicalization overhead (inputs are finite).
__device__ __forceinline__ float vmin(float a, float b) {
    float r;
    asm("v_min_num_f32 %0, %1, %2" : "=v"(r) : "v"(a), "v"(b));
    return r;
}
__device__ __forceinline__ float vmin3(float a, float b, float c) {
    float r;
    asm("v_min3_num_f32 %0, %1, %2, %3" : "=v"(r) : "v"(a), "v"(b), "v"(c));
    return r;
}

#define ROWT 4          // 16-row tiles per wave  (64 rows/wave, 512 rows/block)
#define CSLICES 8       // column slices (grid.z) for device occupancy

// Pre-swizzle the augmented B operand into the exact WMMA lane layout:
// lane<16 : { -2*y0, -2*y1 }   lane>=16 : { -2*y2, |y|^2 }
// bswz[b][chunk][lane], chunk = 16 columns.
__global__ void build_bswz_kernel(const float* __restrict__ y,
                                  v2f* __restrict__ bswz, int M, int total)
{
    int idx = blockIdx.x * 256 + threadIdx.x;
    if (idx >= total) return;
    const int lane  = idx & 31;
    const int chunk = (idx >> 5) % (M >> 4);
    const int b     = idx / ((M >> 4) << 5);
    const int half  = lane >> 4;
    const int col   = (chunk << 4) + (lane & 15);
    const float* yp = y + ((size_t)b * M + col) * 3;
    const float y0 = yp[0], y1 = yp[1], y2 = yp[2];
    v2f v;
    v.x = half ? (-2.0f * y2) : (-2.0f * y0);
    v.y = half ? (y0 * y0 + y1 * y1 + y2 * y2) : (-2.0f * y1);
    bswz[idx] = v;
}

__global__ __launch_bounds__(256) void chamfer_tiles_kernel(
    const float* __restrict__ x, const v2f* __restrict__ bswz,
    float* __restrict__ per_x, float* __restrict__ per_y,
    int N, int M)
{
    __shared__ float wcol[2][8][32];   // double-buffered column mins

    const int b    = blockIdx.y;
    const int tid  = threadIdx.x;
    const int wave = tid >> 5;
    const int lane = tid & 31;
    const int half = lane >> 4;
    const int m16  = lane & 15;

    const int rowBase = (blockIdx.x * 8 + wave) * (ROWT * 16);

    // ---- loop-invariant A tiles (x rows) + C tiles (|x|^2, C/D layout) ----
    v2f a[ROWT]; v8f c[ROWT]; v8f rmin[ROWT];
#pragma unroll
    for (int t = 0; t < ROWT; ++t) {
        const int row   = rowBase + t * 16 + m16;
        const float* xp = x + ((size_t)b * N + row) * 3;
        const float x0 = xp[0], x1 = xp[1], x2 = xp[2];
        const float xn = x0 * x0 + x1 * x1 + x2 * x2;
        a[t].x = half ? x2   : x0;
        a[t].y = half ? 1.0f : x1;
#pragma unroll
        for (int r = 0; r < 8; ++r) {
            c[t][r]    = __shfl(xn, (half << 3) + r, 32);
            rmin[t][r] = __builtin_inff();
        }
    }

    // this block's column-slice: 2 chunks (32 columns) per iteration
    const int chunksTotal = M >> 4;
    const int chunksSlice = chunksTotal / CSLICES;
    const int chunk0      = blockIdx.z * chunksSlice;
    const int nIter       = chunksSlice >> 1;

    const v2f* bbase = bswz + ((size_t)b * chunksTotal + chunk0) * 32;
    v2f b0 = bbase[lane];
    v2f b1 = bbase[32 + lane];

    for (int j = 0; j < nIter; ++j) {
        // software prefetch of next iteration's B (lands behind 8 WMMAs)
        v2f nb0 = b0, nb1 = b1;
        if (j + 1 < nIter) {
            const v2f* np = bbase + (size_t)(j + 1) * 64;
            nb0 = np[lane];
            nb1 = np[32 + lane];
        }

        float colA = 0.0f, colB = 0.0f;
#pragma unroll
        for (int t = 0; t < ROWT; ++t) {
            v8f dA = __builtin_amdgcn_wmma_f32_16x16x4_f32(
                false, a[t], false, b0, (short)0, c[t], false, false);
            v8f dB = __builtin_amdgcn_wmma_f32_16x16x4_f32(
                false, a[t], false, b1, (short)0, c[t], false, false);
#pragma unroll
            for (int r = 0; r < 8; ++r)                 // per-row running min
                rmin[t][r] = vmin3(rmin[t][r], dA[r], dB[r]);
            float tA = vmin3(dA[0], dA[1], dA[2]);      // per-column min trees
            tA = vmin3(tA, dA[3], dA[4]);
            tA = vmin3(tA, dA[5], dA[6]);
            tA = vmin(tA, dA[7]);
            float tB = vmin3(dB[0], dB[1], dB[2]);
            tB = vmin3(tB, dB[3], dB[4]);
            tB = vmin3(tB, dB[5], dB[6]);
            tB = vmin(tB, dB[7]);
            colA = t ? vmin(colA, tA) : tA;
            colB = t ? vmin(colB, tB) : tB;
        }
        // combine lane-halves (rows 0-7 vs 8-15 of each tile)
        colA = vmin(colA, __shfl_xor(colA, 16, 32));
        colB = vmin(colB, __shfl_xor(colB, 16, 32));

        const int buf = j & 1;
        if (half == 0) {
            wcol[buf][wave][m16]      = colA;
            wcol[buf][wave][16 + m16] = colB;
        }
        __syncthreads();
        if (tid < 32) {   // wave 0: fold 8 waves (512 rows) -> 1 atomic/col
            float v = vmin3(wcol[buf][0][tid], wcol[buf][1][tid], wcol[buf][2][tid]);
            v = vmin3(v, wcol[buf][3][tid], wcol[buf][4][tid]);
            v = vmin3(v, wcol[buf][5][tid], wcol[buf][6][tid]);
            v = vmin(v, wcol[buf][7][tid]);
            atomicMin(&per_y[(size_t)b * M + ((chunk0 + j * 2) << 4) + tid], v);
        }
        b0 = nb0;
        b1 = nb1;
    }

    // per_x: min across 16 lanes of each half, then atomic over column slices
#pragma unroll
    for (int off = 1; off < 16; off <<= 1)
#pragma unroll
        for (int t = 0; t < ROWT; ++t)
#pragma unroll
            for (int r = 0; r < 8; ++r)
                rmin[t][r] = vmin(rmin[t][r], __shfl_xor(rmin[t][r], off, 32));
    if (m16 == 0) {
#pragma unroll
        for (int t = 0; t < ROWT; ++t)
#pragma unroll
            for (int r = 0; r < 8; ++r)
                atomicMin(&per_x[(size_t)b * N + rowBase + t * 16 + (half << 3) + r],
                          rmin[t][r]);
    }
}

__global__ void fill_inf_kernel(float* __restrict__ p, int n)
{
    int i = blockIdx.x * 256 + threadIdx.x;
    if (i < n) p[i] = __builtin_inff();
}

__global__ __launch_bounds__(256) void chamfer_reduce_kernel(
    const float* __restrict__ per_x, const float* __restrict__ per_y,
    float* __restrict__ out, int nx, int ny)
{
    __shared__ float s[256];
    const int tid = threadIdx.x;
    float sx = 0.0f, sy = 0.0f;
    for (int i = tid; i < nx; i += 256) sx += per_x[i];
    for (int i = tid; i < ny; i += 256) sy += per_y[i];
    s[tid] = sx / (float)nx + sy / (float)ny;   // linear: tree-sum == mean+mean
    __syncthreads();
    for (int stride = 128; stride > 0; stride >>= 1) {
        if (tid < stride) s[tid] += s[tid + stride];
        __syncthreads();
    }
    if (tid == 0) out[0] = s[0];
}

extern "C" void kernel_launch(void* const* d_in, const int* in_sizes, int n_in,
                              void* d_out, int out_size, void* d_ws, size_t ws_size,
                              hipStream_t stream)
{
    const float* x = (const float*)d_in[0];   // [B,N,3] fp32
    const float* y = (const float*)d_in[1];   // [B,M,3] fp32
    const int B = 8;                          // per reference setup
    const int N = in_sizes[0] / (B * 3);      // 8192
    const int M = in_sizes[1] / (B * 3);      // 8192

    float* per_x = (float*)d_ws;              // [B,N]
    float* per_y = per_x + (size_t)B * N;     // [B,M]
    v2f*   bswz  = (v2f*)(per_y + (size_t)B * M);   // [B][M/16][32]
    float* out   = (float*)d_out;

    const int ninit = B * (N + M);            // per_x and per_y are contiguous
    fill_inf_kernel<<<(ninit + 255) / 256, 256, 0, stream>>>(per_x, ninit);

    const int btot = B * (M >> 4) * 32;
    build_bswz_kernel<<<(btot + 255) / 256, 256, 0, stream>>>(y, bswz, M, btot);

    dim3 grid(N / (8 * ROWT * 16), B, CSLICES);   // 16 x 8 x 8 = 1024 blocks
    chamfer_tiles_kernel<<<grid, 256, 0, stream>>>(x, bswz, per_x, per_y, N, M);

    chamfer_reduce_kernel<<<1, 256, 0, stream>>>(per_x, per_y, out, B * N, B * M);
}